// IlluminationPeakModel_38044820308442
// MI455X (gfx1250) — compile-verified
//
#include <hip/hip_runtime.h>
#include <hip/hip_bf16.h>

typedef __attribute__((ext_vector_type(2))) float v2f;
typedef __attribute__((ext_vector_type(8))) float v8f;

#define N_TBINS 1024
#define PEAK_FACTOR 5.0f

// ---------------------------------------------------------------------------
// Kernel 1: precompute H[1024] = circ_conv(min(circ_conv(relu(L),irf)/area, 5), irf)
// One block, 1024 threads, direct O(T^2) circular convolutions in LDS.
// ---------------------------------------------------------------------------
__global__ __launch_bounds__(1024) void ipm_precompute_H(
    const float* __restrict__ learn, const float* __restrict__ irf,
    float* __restrict__ Hout)
{
    __shared__ float Ls[N_TBINS];   // relu(learnable), later h[t]
    __shared__ float Fs[N_TBINS];   // irf
    __shared__ float Ps[N_TBINS];   // inp = circ_conv(relu(L), irf)
    __shared__ float red[N_TBINS];  // reduction scratch

    const int t = threadIdx.x;
    Ls[t] = fmaxf(learn[t], 0.0f);
    Fs[t] = irf[t];
    __syncthreads();

    // inp[t] = sum_s L[s] * irf[(t - s) mod T]
    float acc = 0.0f;
    for (int s = 0; s < N_TBINS; ++s)
        acc = fmaf(Ls[s], Fs[(t - s) & (N_TBINS - 1)], acc);
    Ps[t]  = acc;
    red[t] = acc;
    __syncthreads();

    // area = sum(inp), deterministic tree reduction
    for (int off = N_TBINS / 2; off > 0; off >>= 1) {
        if (t < off) red[t] += red[t + off];
        __syncthreads();
    }
    const float area = red[0];
    __syncthreads();

    // h[t] = min(inp[t] / area, PEAK_FACTOR)   (batch-independent clamp)
    Ls[t] = fminf(Ps[t] / area, PEAK_FACTOR);
    __syncthreads();

    // H[t] = sum_s h[s] * irf[(t - s) mod T]
    float acc2 = 0.0f;
    for (int s = 0; s < N_TBINS; ++s)
        acc2 = fmaf(Ls[s], Fs[(t - s) & (N_TBINS - 1)], acc2);
    Hout[t] = acc2;
}

// ---------------------------------------------------------------------------
// Kernel 2: two waves cooperate on one 16(batch) x 16(codes) output tile,
// splitting the T=1024 accumulation in half (4096 waves total for MLP).
// shifted[b,t] = pc[b]*H[(t-s_b)&1023] + amb[b]
// noisy        = shifted + sqrt(shifted)*noise[b,t]
// out[b,k]     = sum_t noisy[b,t]*cmat[t,k]  via V_WMMA_F32_16X16X4_F32
// Block: 256 threads = 8 waves = 4 tiles.
// ---------------------------------------------------------------------------
__global__ __launch_bounds__(256) void ipm_project(
    const float* __restrict__ Hws,   const float* __restrict__ cmat,
    const float* __restrict__ noise, const float* __restrict__ pc,
    const float* __restrict__ sbr,   const int*  __restrict__ bins,
    float* __restrict__ out)
{
    __shared__ float ldsH2[2 * N_TBINS];   // 8 KB: doubled table -> linear gather
    __shared__ float partial[4][256];      // 4 KB: cross-wave accumulator combine

    const int tid = threadIdx.x;
    for (int i = tid; i < 2 * N_TBINS; i += 256)
        ldsH2[i] = Hws[i & (N_TBINS - 1)];
    __syncthreads();

    const int wave  = tid >> 5;          // 0..7
    const int lane  = tid & 31;
    const int m     = lane & 15;         // batch row within tile / output col N
    const int khalf = lane >> 4;         // 0: K=0..1, 1: K=2..3 (A and B identical)
    const int pairI = wave >> 1;         // 0..3: tile within block
    const int half  = wave & 1;          // T-range half

    const int tileB = blockIdx.x * 4 + pairI;
    const int row   = tileB * 16 + m;

    const float pcv  = pc[row];
    const float ambv = pcv / sbr[row] * (1.0f / (float)N_TBINS);
    const int   sm   = bins[row] & (N_TBINS - 1);

    // linear gather base: ldsH2[1024 - sm + k] == H[(k - sm) & 1023], k in [0,1023]
    const float* hbase = ldsH2 + (N_TBINS - sm) + khalf * 2;
    const float* nrow  = noise + (size_t)row * N_TBINS + khalf * 2;  // b64 stream
    const float* cbase = cmat + khalf * 2 * 16 + m;                  // B fragment col

    const int tlo = half * (N_TBINS / 2);
    const int thi = tlo + (N_TBINS / 2);

    v8f acc = {};   // 16x16 fp32 accumulator

    for (int tb = tlo; tb < thi; tb += 32) {
        // stream-ahead hint for this lane's noise row (~2KB ahead, speculative)
        __builtin_prefetch(nrow + tb + 512, 0, 1);
#pragma unroll
        for (int u = 0; u < 32; u += 4) {
            const int t0 = tb + u;

            // noise pair (aligned b64, non-temporal: read-once 128MB stream)
            const v2f nz = __builtin_nontemporal_load((const v2f*)(nrow + t0));

            // template gather: immediate-offset LDS reads, no wrap math
            const float h0 = hbase[t0];
            const float h1 = hbase[t0 + 1];

            // B fragment: cmat rows k0,k0+1 col m (64B-coalesced, cache-resident)
            const float c0 = cbase[t0 * 16];
            const float c1 = cbase[t0 * 16 + 16];

            // shifted -> noisy (raw v_sqrt_f32: args are O(10..5000), no denorms)
            const float s0 = fmaf(pcv, h0, ambv);
            const float s1 = fmaf(pcv, h1, ambv);
            const float a0 = fmaf(__builtin_amdgcn_sqrtf(s0), nz.x, s0);
            const float a1 = fmaf(__builtin_amdgcn_sqrtf(s1), nz.y, s1);

            const v2f A = {a0, a1};
            const v2f B = {c0, c1};
            // D = A(16x4) * B(4x16) + C, fp32
            acc = __builtin_amdgcn_wmma_f32_16x16x4_f32(
                false, A, false, B, (short)0, acc, false, false);
        }
    }

    // Combine the two T-halves, then store.
    // C/D layout: VGPR r -> rows r (lanes 0-15) and r+8 (lanes 16-31), N = lane&15
    if (half == 0) {
#pragma unroll
        for (int r = 0; r < 8; ++r)
            partial[pairI][(r + khalf * 8) * 16 + m] = acc[r];
    }
    __syncthreads();
    if (half == 1) {
        float* obase = out + (size_t)tileB * 16 * 16;
#pragma unroll
        for (int r = 0; r < 8; ++r) {
            const int M = r + khalf * 8;
            __builtin_nontemporal_store(acc[r] + partial[pairI][M * 16 + m],
                                        obase + M * 16 + m);
        }
    }
}

extern "C" void kernel_launch(void* const* d_in, const int* in_sizes, int n_in,
                              void* d_out, int out_size, void* d_ws, size_t ws_size,
                              hipStream_t stream) {
    const float* learn = (const float*)d_in[0];   // [1024,1]
    const float* irf   = (const float*)d_in[1];   // [1024]
    const float* cmat  = (const float*)d_in[2];   // [1024,16]
    const float* noise = (const float*)d_in[3];   // [B,1024]
    const float* pc    = (const float*)d_in[4];   // [B]
    const float* sbr   = (const float*)d_in[5];   // [B]
    const int*   bins  = (const int*)d_in[6];     // [B]
    float*       outp  = (float*)d_out;           // [B,16]

    float* Hws = (float*)d_ws;                    // 1024 floats of scratch

    const int batch  = in_sizes[4];
    const int tiles  = batch / 16;                // 2048
    const int blocks = tiles / 4;                 // 4 tiles (8 waves) per block

    ipm_precompute_H<<<1, 1024, 0, stream>>>(learn, irf, Hws);
    ipm_project<<<blocks, 256, 0, stream>>>(Hws, cmat, noise, pc, sbr, bins, outp);
}